// MEFSSIM_4724464025885
// MI455X (gfx1250) — compile-verified
//
#include <hip/hip_runtime.h>

typedef __attribute__((ext_vector_type(2))) float v2f;
typedef __attribute__((ext_vector_type(8))) float v8f;

#define IMG 2048
#define NPIX (2048*2048)
#define NTILE 16384           /* 128x128 tiles of 16x16 */
#define C1_ 0.0001f           /* 0.01^2 */
#define C2_ 0.0009f           /* 0.03^2 */
#define DENOM_ 0.08f          /* 2*0.2^2 */

__device__ __forceinline__ v8f zero8() {
  v8f z = {0.f,0.f,0.f,0.f,0.f,0.f,0.f,0.f};
  return z;
}

// Branch-free shared-load sample: one address computation + one load set yields
// all plane values of a group. NP==2: {SX, SX2} from X. NP==3: {SY, SY2, SXY}
// from exposure plane Yk (pre-offset) + X. Zero outside the image (SAME pad).
template<int NP>
__device__ __forceinline__ void sample_group(const float* __restrict__ X,
                                             const float* __restrict__ Yk,
                                             size_t rowbase, bool rowok, int x,
                                             float (&out)[NP]) {
  bool ok = rowok && ((unsigned)x < (unsigned)IMG);
  size_t idx = rowbase + (size_t)(ok ? x : 0);   // clamped: always valid
  const size_t CS = (size_t)NPIX;
  float a0 = X[idx], a1 = X[idx + CS], a2 = X[idx + 2*CS];
  if (NP == 2) {
    out[0] = ok ? (a0 + a1 + a2) : 0.f;
    out[1] = ok ? (a0*a0 + a1*a1 + a2*a2) : 0.f;
  } else {
    float b0 = Yk[idx], b1 = Yk[idx + CS], b2 = Yk[idx + 2*CS];
    out[0] = ok ? (b0 + b1 + b2) : 0.f;
    out[1] = ok ? (b0*b0 + b1*b1 + b2*b2) : 0.f;
    out[2] = ok ? (a0*b0 + a1*b1 + a2*b2) : 0.f;
  }
}

// Separable 11-tap conv of NP planes (sharing loads) for one 16x16 tile via
// f32 WMMA band-GEMMs: H(32x16) = In(32x28) x Bh(28x16), Out = Av(16x28) x H.
template<int NP>
__device__ __forceinline__ void conv_group(const float* __restrict__ X,
                                           const float* __restrict__ Yk,
                                           float (*__restrict__ Hsw)[32][17],
                                           int r0, int c0, int nl, bool hi,
                                           const v2f (&Bh)[7], const v2f (&Av)[7],
                                           v8f (&outv)[NP]) {
  #pragma unroll
  for (int b = 0; b < 2; ++b) {
    v8f hacc[NP];
    #pragma unroll
    for (int np = 0; np < NP; ++np) hacc[np] = zero8();
    int y = r0 - 5 + b*16 + nl;
    bool rowok = (unsigned)y < (unsigned)IMG;
    size_t rowbase = (size_t)(rowok ? y : 0) * IMG;
    #pragma unroll
    for (int q = 0; q < 7; ++q) {
      int xb = c0 - 5 + 4*q + (hi ? 2 : 0);
      float s0[NP], s1[NP];
      sample_group<NP>(X, Yk, rowbase, rowok, xb,     s0);
      sample_group<NP>(X, Yk, rowbase, rowok, xb + 1, s1);
      #pragma unroll
      for (int np = 0; np < NP; ++np) {
        v2f a; a.x = s0[np]; a.y = s1[np];
        hacc[np] = __builtin_amdgcn_wmma_f32_16x16x4_f32(
            false, a, false, Bh[q], (short)0, hacc[np], false, false);
      }
    }
    // D layout -> LDS (row M = v (lanes0-15) / v+8 (lanes16-31), col = lane%16)
    #pragma unroll
    for (int np = 0; np < NP; ++np) {
      #pragma unroll
      for (int v = 0; v < 8; ++v)
        Hsw[np][b*16 + v + (hi ? 8 : 0)][nl] = hacc[np][v];
    }
  }
  asm volatile("s_wait_dscnt 0" ::: "memory");   // RAW: transpose visible wave-wide

  #pragma unroll
  for (int np = 0; np < NP; ++np) outv[np] = zero8();
  #pragma unroll
  for (int q = 0; q < 7; ++q) {
    int k0 = 4*q + (hi ? 2 : 0);
    #pragma unroll
    for (int np = 0; np < NP; ++np) {
      v2f bf;
      bf.x = Hsw[np][k0][nl];
      bf.y = Hsw[np][k0 + 1][nl];
      outv[np] = __builtin_amdgcn_wmma_f32_16x16x4_f32(
          false, Av[q], false, bf, (short)0, outv[np], false, false);
    }
  }
  asm volatile("" ::: "memory");                 // WAR: next group's stores stay after these loads
}

__global__ __launch_bounds__(256) void mef_conv_kernel(
    const float* __restrict__ X, const float* __restrict__ Ys,
    float* __restrict__ csm, float* __restrict__ muXm,
    float* __restrict__ muYm, float* __restrict__ lYpart) {
  __shared__ float Hs[8][3][32][17];             // 52,224 B
  const int lane = threadIdx.x & 31;
  const int w    = threadIdx.x >> 5;
  const int nl   = lane & 15;
  const bool hi  = lane >= 16;
  const int gt   = blockIdx.x * 8 + w;           // global tile id
  const int r0   = (gt >> 7) << 4;
  const int c0   = (gt & 127) << 4;
  float (*Hsw)[32][17] = Hs[w];

  // 11-tap Gaussian, sigma = 11/6, normalized (matches _make_window)
  float g[11]; float gs = 0.f;
  #pragma unroll
  for (int t = 0; t < 11; ++t) {
    float d = (float)(t - 5);
    g[t] = expf(-d*d / (2.f * (11.f/6.f) * (11.f/6.f)));
    gs += g[t];
  }
  #pragma unroll
  for (int t = 0; t < 11; ++t) g[t] /= gs;

  // Band-matrix WMMA fragments.
  //  Horizontal B[k][n] = g[k-n]/3 (k-n in [0,10]); vertical A[m][k] = g[k-m].
  //  f32 16x16x4: VGPR0 = K=k0 (lanes0-15)/K=k0+2 (lanes16-31); VGPR1 = k0+1/k0+3.
  v2f Bh[7], Av[7];
  #pragma unroll
  for (int q = 0; q < 7; ++q) {
    int k0 = 4*q + (hi ? 2 : 0);
    int d0 = k0 - nl, d1 = k0 + 1 - nl;
    float b0 = (d0 >= 0 && d0 < 11) ? g[d0] : 0.f;
    float b1 = (d1 >= 0 && d1 < 11) ? g[d1] : 0.f;
    Bh[q].x = b0 * (1.f/3.f); Bh[q].y = b1 * (1.f/3.f);
    Av[q].x = b0;             Av[q].y = b1;
  }

  // Pixel base for this lane: (r0 + v + hi*8, c0 + nl)
  const size_t pix0 = (size_t)(r0 + (hi ? 8 : 0)) * IMG + (size_t)(c0 + nl);

  // Group 0: muX and E[X^2] share X loads.
  v8f xg[2];
  conv_group<2>(X, X, Hsw, r0, c0, nl, hi, Bh, Av, xg);
  float sX2[8];
  #pragma unroll
  for (int v = 0; v < 8; ++v) sX2[v] = xg[1][v] - xg[0][v]*xg[0][v];

  float best[8], cssel[8];
  #pragma unroll
  for (int v = 0; v < 8; ++v) { best[v] = -1e30f; cssel[v] = 0.f; }
  float tsum[4];

  // Groups 1..4: {muY_k, E[Y^2]_k, E[XY]_k} share loads; fold per-pixel math
  // immediately so only muX/sX2/best/cssel stay live.
  #pragma unroll
  for (int k = 0; k < 4; ++k) {
    const float* Yk = Ys + (size_t)k * 3 * NPIX;
    v8f yg[3];
    conv_group<3>(X, Yk, Hsw, r0, c0, nl, hi, Bh, Av, yg);
    float sk = 0.f;
    #pragma unroll
    for (int v = 0; v < 8; ++v) {
      float muY = yg[0][v];
      float sY2 = yg[1][v] - muY*muY;
      float sXY = yg[2][v] - xg[0][v]*muY;
      float cs  = (2.f*sXY + C2_) / (sX2[v] + sY2 + C2_);
      if (sY2 > best[v]) { best[v] = sY2; cssel[v] = cs; }  // first-max == jnp.argmax
      muYm[pix0 + (size_t)v*IMG + (size_t)k*NPIX] = muY;
      sk += muY;
    }
    tsum[k] = sk;
  }

  #pragma unroll
  for (int v = 0; v < 8; ++v) {
    csm[pix0 + (size_t)v*IMG]  = cssel[v];
    muXm[pix0 + (size_t)v*IMG] = xg[0][v];
  }

  // wave32 reduction of tile muY sums (for lY means)
  #pragma unroll
  for (int off = 16; off > 0; off >>= 1) {
    #pragma unroll
    for (int k = 0; k < 4; ++k) tsum[k] += __shfl_xor(tsum[k], off);
  }
  if (lane == 0) {
    #pragma unroll
    for (int k = 0; k < 4; ++k) lYpart[(size_t)gt*4 + k] = tsum[k];
  }
}

__global__ __launch_bounds__(256) void mef_lg_kernel(const float* __restrict__ lYpart,
                                                     float* __restrict__ lG) {
  __shared__ float red[256];
  int t = threadIdx.x;
  for (int k = 0; k < 4; ++k) {
    float s = 0.f;
    for (int i = t; i < NTILE; i += 256) s += lYpart[(size_t)i*4 + k];
    red[t] = s; __syncthreads();
    for (int off = 128; off > 0; off >>= 1) {
      if (t < off) red[t] += red[t + off];
      __syncthreads();
    }
    if (t == 0) {
      float mean = red[0] / (float)NPIX;
      float d = mean - 0.5f;
      lG[k] = expf(-d*d / DENOM_);
    }
    __syncthreads();
  }
}

__global__ __launch_bounds__(256) void mef_fuse_kernel(
    const float* __restrict__ csm, const float* __restrict__ muXm,
    const float* __restrict__ muYm, const float* __restrict__ lG,
    float* __restrict__ qpart) {
  float g0 = lG[0], g1 = lG[1], g2 = lG[2], g3 = lG[3];
  float acc = 0.f;
  size_t base = (size_t)blockIdx.x * 4096 + threadIdx.x;
  for (int i = 0; i < 16; ++i) {
    size_t pix = base + (size_t)i * 256;
    float muX = muXm[pix];
    float cs  = csm[pix];
    float num = 0.f, den = 0.f;
    #pragma unroll
    for (int k = 0; k < 4; ++k) {
      float muY = muYm[pix + (size_t)k * NPIX];
      float d = muY - 0.5f;
      float lw = ((k==0) ? g0 : (k==1) ? g1 : (k==2) ? g2 : g3) * expf(-d*d / DENOM_);
      num += lw * muY; den += lw;
    }
    float muYw = num / den;
    float l = (2.f * muX * muYw + C1_) / (muX * muX + muYw * muYw + C1_);
    acc += l * cs;
  }
  __shared__ float red[256];
  red[threadIdx.x] = acc; __syncthreads();
  for (int off = 128; off > 0; off >>= 1) {
    if (threadIdx.x < off) red[threadIdx.x] += red[threadIdx.x + off];
    __syncthreads();
  }
  if (threadIdx.x == 0) qpart[blockIdx.x] = red[0];
}

__global__ __launch_bounds__(256) void mef_final_kernel(const float* __restrict__ qpart,
                                                        float* __restrict__ out) {
  __shared__ float red[256];
  int t = threadIdx.x;
  float s = qpart[t] + qpart[t + 256] + qpart[t + 512] + qpart[t + 768];
  red[t] = s; __syncthreads();
  for (int off = 128; off > 0; off >>= 1) {
    if (t < off) red[t] += red[t + off];
    __syncthreads();
  }
  if (t == 0) out[0] = red[0] / (float)NPIX;
}

extern "C" void kernel_launch(void* const* d_in, const int* in_sizes, int n_in,
                              void* d_out, int out_size, void* d_ws, size_t ws_size,
                              hipStream_t stream) {
  const float* X  = (const float*)d_in[0];   // [1,3,2048,2048]
  const float* Ys = (const float*)d_in[1];   // [4,3,2048,2048]
  float* ws    = (float*)d_ws;
  float* csm   = ws;                           // [NPIX]
  float* muXm  = ws + (size_t)NPIX;            // [NPIX]
  float* muYm  = ws + 2*(size_t)NPIX;          // [4*NPIX]
  float* lYpart= ws + 6*(size_t)NPIX;          // [NTILE*4]
  float* lG    = lYpart + (size_t)NTILE*4;     // [4]
  float* qpart = lG + 4;                       // [1024]

  mef_conv_kernel<<<2048, 256, 0, stream>>>(X, Ys, csm, muXm, muYm, lYpart);
  mef_lg_kernel<<<1, 256, 0, stream>>>(lYpart, lG);
  mef_fuse_kernel<<<1024, 256, 0, stream>>>(csm, muXm, muYm, lG, qpart);
  mef_final_kernel<<<1, 256, 0, stream>>>(qpart, (float*)d_out);
}